// GameNNUE_12695923327232
// MI455X (gfx1250) — compile-verified
//
#include <hip/hip_runtime.h>
#include <hip/hip_bf16.h>

#define FEAT  40960
#define ACC   512
#define NBATCH 4096
#define L1N   32
#define MAXNNZ 1024

typedef float v2f __attribute__((ext_vector_type(2)));
typedef float v4f __attribute__((ext_vector_type(4)));
typedef float v8f __attribute__((ext_vector_type(8)));

// ---------------------------------------------------------------------------
// Kernel A: sparse feature transform.
// One workgroup per (batch row, side). Phase 1: stream the 160KB of 0/1
// feature floats (non-temporal, keep ft_w hot in L2), append nonzero indices
// to LDS. Phase 2: each of 256 threads owns 2 of the 512 accumulator columns,
// sums ft_w rows (features are exactly 1.0 so row-add == row*feat), applies
// screlu, and writes to the stm-selected half of the concatenated x buffer.
// ---------------------------------------------------------------------------
__global__ __launch_bounds__(256) void ft_kernel(
    const float* __restrict__ wf, const float* __restrict__ bf,
    const unsigned char* __restrict__ stm,
    const float* __restrict__ ft_w, const float* __restrict__ ft_b,
    float* __restrict__ x)
{
    __shared__ int s_idx[MAXNNZ];
    __shared__ int s_cnt;

    const int wg   = blockIdx.x;      // 0 .. 2*NBATCH-1
    const int row  = wg >> 1;
    const int side = wg & 1;          // 0 = white, 1 = black
    const int tid  = threadIdx.x;

    if (tid == 0) s_cnt = 0;
    __syncthreads();

    const float* feat = (side ? bf : wf) + (size_t)row * FEAT;
    const v4f* f4 = (const v4f*)feat;

    // FEAT/4 = 10240 vec4s; 256 threads -> 40 iterations, fully coalesced.
    #pragma unroll 4
    for (int it = 0; it < FEAT / 4 / 256; ++it) {
        const int vidx = tid + it * 256;
        v4f v = __builtin_nontemporal_load(&f4[vidx]);
        const int base = vidx * 4;
        if (v.x != 0.0f) { int p = atomicAdd(&s_cnt, 1); if (p < MAXNNZ) s_idx[p] = base;     }
        if (v.y != 0.0f) { int p = atomicAdd(&s_cnt, 1); if (p < MAXNNZ) s_idx[p] = base + 1; }
        if (v.z != 0.0f) { int p = atomicAdd(&s_cnt, 1); if (p < MAXNNZ) s_idx[p] = base + 2; }
        if (v.w != 0.0f) { int p = atomicAdd(&s_cnt, 1); if (p < MAXNNZ) s_idx[p] = base + 3; }
    }
    __syncthreads();

    int nnz = s_cnt;
    if (nnz > MAXNNZ) nnz = MAXNNZ;

    // Accumulate: thread owns columns {2*tid, 2*tid+1}. ft_w rows are 2KB,
    // all hits in the 192MB L2 (ft_w = 80MB, protected by NT feature loads).
    const int c0 = tid * 2;
    v2f acc = *(const v2f*)(ft_b + c0);
    for (int i = 0; i < nnz; ++i) {
        const int f = s_idx[i];                       // LDS broadcast
        v2f w = *(const v2f*)(ft_w + (size_t)f * ACC + c0);
        acc.x += w.x;
        acc.y += w.y;
    }

    // screlu
    acc.x = fminf(fmaxf(acc.x, 0.0f), 1.0f); acc.x *= acc.x;
    acc.y = fminf(fmaxf(acc.y, 0.0f), 1.0f); acc.y *= acc.y;

    // stm select: side's accum goes to first half iff side == stm(row)
    const int m    = stm[row] ? 1 : 0;
    const int half = (side == m) ? 0 : 1;
    *(v2f*)(x + (size_t)row * (2 * ACC) + half * ACC + c0) = acc;
}

// ---------------------------------------------------------------------------
// Kernel B: MLP tail with fp32 WMMA (exact precision vs f32 reference).
// One workgroup = 16 batch rows, 64 threads = 2 waves (wave32).
// Wave w computes output columns [16w, 16w+16) of each 16x32 layer output.
//
// V_WMMA_F32_16X16X4_F32 operand layout (ISA 7.12.2):
//   A (16x4):  lane 0-15 -> M=lane, VGPR0=K0, VGPR1=K1 ; lane 16-31 -> K2,K3
//   B (4x16):  lane 0-15 -> N=lane, VGPR0=K0, VGPR1=K1 ; lane 16-31 -> K2,K3
//   C/D 16x16: VGPR r, lanes 0-15 -> M=r, N=lane ; lanes 16-31 -> M=r+8
// ---------------------------------------------------------------------------
__global__ __launch_bounds__(64) void tail_kernel(
    const float* __restrict__ x,
    const float* __restrict__ l1_w, const float* __restrict__ l1_b,
    const float* __restrict__ l2_w, const float* __restrict__ l2_b,
    const float* __restrict__ out_w, const float* __restrict__ out_b,
    float* __restrict__ out)
{
    __shared__ float s_x[16 * 1024];   // 64 KB activation tile
    __shared__ float s_v[16 * L1N];    // L1 output (screlu'd)
    __shared__ float s_v2[16 * L1N];   // L2 output (screlu'd)

    const int row0 = blockIdx.x * 16;
    const int tid  = threadIdx.x;      // 0..63
    const int lane = tid & 31;
    const int wave = tid >> 5;         // 0 or 1

    // Cooperative tile load: 4096 float4s across 64 threads.
    const v4f* xs = (const v4f*)(x + (size_t)row0 * 1024);
    v4f* sd = (v4f*)s_x;
    for (int i = tid; i < 16 * 1024 / 4; i += 64) sd[i] = xs[i];
    __syncthreads();

    const int noff  = wave * 16;
    const int mrow  = lane & 15;       // A-frag row / B-frag column-in-tile
    const int khalf = (lane >> 4) * 2; // 0 or 2: which K pair this half-wave holds
    const int ncol  = noff + mrow;     // this lane's output column

    // ---- L1: [16 x 1024] @ [1024 x 16]  (256 k-steps of 16x16x4) ----
    v8f c = {};
    for (int k = 0; k < 1024; k += 4) {
        v2f a = *(const v2f*)(&s_x[mrow * 1024 + k + khalf]);
        v2f b = *(const v2f*)(l1_w + (size_t)ncol * 1024 + k + khalf); // B[k][n]=l1_w[n][k]
        c = __builtin_amdgcn_wmma_f32_16x16x4_f32(
                false, a, false, b, (short)0, c, false, false);
    }
    {
        const float bias = l1_b[ncol];
        #pragma unroll
        for (int r = 0; r < 8; ++r) {
            const int mm = r + 8 * (lane >> 4);
            float v = c[r] + bias;
            v = fminf(fmaxf(v, 0.0f), 1.0f); v *= v;
            s_v[mm * L1N + ncol] = v;
        }
    }
    __syncthreads();

    // ---- L2: [16 x 32] @ [32 x 16]  (8 k-steps) ----
    v8f c2 = {};
    #pragma unroll
    for (int k = 0; k < L1N; k += 4) {
        v2f a = *(const v2f*)(&s_v[mrow * L1N + k + khalf]);
        v2f b = *(const v2f*)(l2_w + (size_t)ncol * L1N + k + khalf);
        c2 = __builtin_amdgcn_wmma_f32_16x16x4_f32(
                false, a, false, b, (short)0, c2, false, false);
    }
    {
        const float bias = l2_b[ncol];
        #pragma unroll
        for (int r = 0; r < 8; ++r) {
            const int mm = r + 8 * (lane >> 4);
            float v = c2[r] + bias;
            v = fminf(fmaxf(v, 0.0f), 1.0f); v *= v;
            s_v2[mm * L1N + ncol] = v;
        }
    }
    __syncthreads();

    // ---- Output layer: 32 -> 1 per row ----
    if (tid < 16) {
        float s = out_b[0];
        #pragma unroll
        for (int n = 0; n < L1N; ++n) s += s_v2[tid * L1N + n] * out_w[n];
        out[row0 + tid] = s;
    }
}

// ---------------------------------------------------------------------------
extern "C" void kernel_launch(void* const* d_in, const int* in_sizes, int n_in,
                              void* d_out, int out_size, void* d_ws, size_t ws_size,
                              hipStream_t stream) {
    const float*         wf    = (const float*)d_in[0];
    const float*         bf    = (const float*)d_in[1];
    const unsigned char* stm   = (const unsigned char*)d_in[2]; // jnp.bool_ -> 1 byte
    const float*         ft_w  = (const float*)d_in[3];
    const float*         ft_b  = (const float*)d_in[4];
    const float*         l1_w  = (const float*)d_in[5];
    const float*         l1_b  = (const float*)d_in[6];
    const float*         l2_w  = (const float*)d_in[7];
    const float*         l2_b  = (const float*)d_in[8];
    const float*         out_w = (const float*)d_in[9];
    const float*         out_b = (const float*)d_in[10];

    float* x = (float*)d_ws;  // [NBATCH][1024] concatenated screlu accums, 16 MB

    ft_kernel<<<NBATCH * 2, 256, 0, stream>>>(wf, bf, stm, ft_w, ft_b, x);
    tail_kernel<<<NBATCH / 16, 64, 0, stream>>>(x, l1_w, l1_b, l2_w, l2_b,
                                                out_w, out_b, (float*)d_out);
}